// CGCNN_15161234555427
// MI455X (gfx1250) — compile-verified
//
#include <hip/hip_runtime.h>
#include <hip/hip_bf16.h>

typedef float v2f __attribute__((ext_vector_type(2)));
typedef float v8f __attribute__((ext_vector_type(8)));

#define NODE_F   64
#define EDGE_F   41
#define DIN      169            // 2*NODE_F + EDGE_F
#define DOUT     128            // 2*NODE_F
#define KCHUNKS  43             // ceil(169/4)
#define FRAGS_PER_LAYER (KCHUNKS * 8 * 32)   // 11008 float2 per layer
#define EPS      1e-5f

// ---------------------------------------------------------------- helpers
__device__ __forceinline__ float softplusf_(float v) {
  // jax.nn.softplus == logaddexp(v, 0) == max(v,0) + log1p(exp(-|v|))
  return fmaxf(v, 0.0f) + log1pf(expf(-fabsf(v)));
}
__device__ __forceinline__ float sigmoidf_(float v) {
  float e = expf(-fabsf(v));
  float d = 1.0f + e;
  return (v >= 0.0f) ? (1.0f / d) : (e / d);
}

// ---------------------------------------------------------------- W pre-swizzle
// Store Wf[l] (169x128, K-major rows) into WMMA B-fragment order:
// frag fi = (c*8 + t)*32 + lane ; kb = 4c + 2*(lane>=16) ; ncol = t*16 + (lane&15)
// value = { W[kb][ncol], W[kb+1][ncol] } with zero padding for kb >= 169.
__global__ void cg_prep_w(const float* __restrict__ Wf, float2* __restrict__ wswz) {
  int total = 3 * FRAGS_PER_LAYER;
  for (int fi = blockIdx.x * blockDim.x + threadIdx.x; fi < total;
       fi += gridDim.x * blockDim.x) {
    int l    = fi / FRAGS_PER_LAYER;
    int r    = fi % FRAGS_PER_LAYER;
    int c    = r / 256;
    int rem  = r % 256;
    int t    = rem / 32;
    int lane = rem % 32;
    int half = lane >> 4;
    int kb   = 4 * c + 2 * half;
    int ncol = t * 16 + (lane & 15);
    const float* base = Wf + (size_t)l * DIN * DOUT;
    float w0 = (kb     < DIN) ? base[(size_t)kb * DOUT + ncol] : 0.0f;
    float w1 = (kb + 1 < DIN) ? base[(size_t)(kb + 1) * DOUT + ncol] : 0.0f;
    float2 v; v.x = w0; v.y = w1;
    wswz[fi] = v;
  }
}

// ---------------------------------------------------------------- tile GEMM core
// One wave computes a 16-edge x 128-col tile of z = [x[i1],x[i2],ef] @ Wf.
// A layout (f32 16x4): lane<16 -> row=lane, K=kb..kb+1 ; lane>=16 -> row=lane-16, K=kb+2..kb+3.
__device__ __forceinline__ void cg_edge_tile(
    const float* __restrict__ x, const float* __restrict__ ef,
    const int* __restrict__ idx1, const int* __restrict__ idx2,
    const float2* __restrict__ wswz, int e0, int lane, v8f acc[8]) {
  const int r    = lane & 15;
  const int half = lane >> 4;
  const int e    = e0 + r;
  const float* px1 = x + (size_t)idx1[e] * NODE_F;
  const float* px2 = x + (size_t)idx2[e] * NODE_F;
  const float* pef = ef + (size_t)e * EDGE_F;
  const float2* pB = wswz + lane;

  for (int c = 0; c < KCHUNKS; ++c) {
    int kb = 4 * c + 2 * half;
    v2f a;
    if (kb < NODE_F) {                       // from x[idx1]
      float2 t2 = *(const float2*)(px1 + kb);
      a.x = t2.x; a.y = t2.y;
    } else if (kb < 2 * NODE_F) {            // from x[idx2]
      float2 t2 = *(const float2*)(px2 + (kb - NODE_F));
      a.x = t2.x; a.y = t2.y;
    } else if (kb < 2 * NODE_F + EDGE_F - 1) { // edge features (4B aligned only)
      a.x = pef[kb - 2 * NODE_F];
      a.y = pef[kb - 2 * NODE_F + 1];
    } else if (kb == 168) {                  // last edge feature + zero pad
      a.x = pef[EDGE_F - 1];
      a.y = 0.0f;
    } else {                                 // full zero pad (kb = 170)
      a.x = 0.0f; a.y = 0.0f;
    }
    const float2* pBc = pB + c * 256;
#pragma unroll
    for (int t = 0; t < 8; ++t) {
      float2 bw = pBc[t * 32];
      v2f b; b.x = bw.x; b.y = bw.y;
      acc[t] = __builtin_amdgcn_wmma_f32_16x16x4_f32(
          false, a, false, b, (short)0, acc[t], false, false);
    }
  }
}

// ---------------------------------------------------------------- pass 1: BN column stats
__global__ void __launch_bounds__(128) cg_gemm_stats(
    const float* __restrict__ x, const float* __restrict__ ef,
    const int* __restrict__ idx1, const int* __restrict__ idx2,
    const float2* __restrict__ wswz, const float* __restrict__ bfl,
    int ntiles, float* __restrict__ zsum, float* __restrict__ zsq) {
  __shared__ float ls[DOUT];
  __shared__ float lq[DOUT];
  int tid = threadIdx.x;
  ls[tid] = 0.0f; lq[tid] = 0.0f;
  __syncthreads();

  int lane = tid & 31, w = tid >> 5;
  int r = lane & 15;
  float bias[8];
#pragma unroll
  for (int t = 0; t < 8; ++t) bias[t] = bfl[t * 16 + r];

  float ssum[8], ssq[8];
#pragma unroll
  for (int t = 0; t < 8; ++t) { ssum[t] = 0.0f; ssq[t] = 0.0f; }

  int gw = blockIdx.x * 4 + w, nw = gridDim.x * 4;
  for (int tile = gw; tile < ntiles; tile += nw) {
    v8f acc[8];
#pragma unroll
    for (int t = 0; t < 8; ++t) acc[t] = (v8f){0,0,0,0,0,0,0,0};
    cg_edge_tile(x, ef, idx1, idx2, wswz, tile * 16, lane, acc);
#pragma unroll
    for (int t = 0; t < 8; ++t)
#pragma unroll
      for (int i = 0; i < 8; ++i) {
        float v = acc[t][i] + bias[t];
        ssum[t] += v; ssq[t] += v * v;
      }
  }
#pragma unroll
  for (int t = 0; t < 8; ++t) {
    atomicAdd(&ls[t * 16 + r], ssum[t]);
    atomicAdd(&lq[t * 16 + r], ssq[t]);
  }
  __syncthreads();
  atomicAdd(&zsum[tid], ls[tid]);
  atomicAdd(&zsq[tid],  lq[tid]);
}

// ---------------------------------------------------------------- BN finalize (128 or 64 cols)
__global__ void cg_bn_final(float* __restrict__ sum, float* __restrict__ sq,
                            const float* __restrict__ g, const float* __restrict__ be,
                            float inv_n, float* __restrict__ s, float* __restrict__ t,
                            int cols) {
  int i = threadIdx.x;
  if (i < cols) {
    float m  = sum[i] * inv_n;
    float v  = sq[i] * inv_n - m * m;      // biased variance
    float sc = g[i] * rsqrtf(v + EPS);
    s[i] = sc;
    t[i] = be[i] - m * sc;
    sum[i] = 0.0f; sq[i] = 0.0f;           // reset for next use
  }
}

// ---------------------------------------------------------------- pass 2: msg + scatter
__global__ void __launch_bounds__(128) cg_msg_scatter(
    const float* __restrict__ x, const float* __restrict__ ef,
    const int* __restrict__ idx1, const int* __restrict__ idx2,
    const float2* __restrict__ wswz, const float* __restrict__ bfl,
    const float* __restrict__ s1, const float* __restrict__ t1,
    float* __restrict__ nodeacc, float* __restrict__ ncount, int ntiles) {
  int tid = threadIdx.x, lane = tid & 31, w = tid >> 5;
  int r = lane & 15, hf = lane >> 4;

  float sc[8], tc[8];
#pragma unroll
  for (int t = 0; t < 8; ++t) {
    int col = t * 16 + r;
    float s = s1[col];
    sc[t] = s;
    tc[t] = t1[col] + bfl[col] * s;   // fold bias into shift
  }

  int gw = blockIdx.x * 4 + w, nw = gridDim.x * 4;
  for (int tile = gw; tile < ntiles; tile += nw) {
    int e0 = tile * 16;
    v8f acc[8];
#pragma unroll
    for (int t = 0; t < 8; ++t) acc[t] = (v8f){0,0,0,0,0,0,0,0};
    cg_edge_tile(x, ef, idx1, idx2, wswz, e0, lane, acc);

    int dest[8];
#pragma unroll
    for (int i = 0; i < 8; ++i) dest[i] = idx1[e0 + hf * 8 + i];
    if (r == 0) {
#pragma unroll
      for (int i = 0; i < 8; ++i) atomicAdd(&ncount[dest[i]], 1.0f);
    }
#pragma unroll
    for (int t = 0; t < 4; ++t) {
      int col = t * 16 + r;                     // gate col 0..63
#pragma unroll
      for (int i = 0; i < 8; ++i) {
        float gte = acc[t][i]     * sc[t]     + tc[t];
        float cnv = acc[t + 4][i] * sc[t + 4] + tc[t + 4];
        float m = sigmoidf_(gte) * softplusf_(cnv);
        atomicAdd(&nodeacc[(size_t)dest[i] * NODE_F + col], m);
      }
    }
  }
}

// ---------------------------------------------------------------- node mean + column stats
__global__ void __launch_bounds__(256) cg_node_stats(
    float* __restrict__ nodeacc, const float* __restrict__ ncount, int total,
    float* __restrict__ nsum, float* __restrict__ nsq) {
  __shared__ float ls[NODE_F];
  __shared__ float lq[NODE_F];
  int tid = threadIdx.x;
  if (tid < NODE_F) { ls[tid] = 0.0f; lq[tid] = 0.0f; }
  __syncthreads();
  int idx = blockIdx.x * 256 + tid, stride = gridDim.x * 256;  // stride % 64 == 0
  int f = idx & (NODE_F - 1);
  float a = 0.0f, q = 0.0f;
  for (int i = idx; i < total; i += stride) {
    int n = i >> 6;
    float mm = nodeacc[i] / fmaxf(ncount[n], 1.0f);
    nodeacc[i] = mm;
    a += mm; q += mm * mm;
  }
  atomicAdd(&ls[f], a); atomicAdd(&lq[f], q);
  __syncthreads();
  if (tid < NODE_F) { atomicAdd(&nsum[tid], ls[tid]); atomicAdd(&nsq[tid], lq[tid]); }
}

// ---------------------------------------------------------------- node update + reset accum
__global__ void cg_node_update(float* __restrict__ x, float* __restrict__ nodeacc,
                               float* __restrict__ ncount, const float* __restrict__ s2,
                               const float* __restrict__ t2, int total) {
  for (int i = blockIdx.x * blockDim.x + threadIdx.x; i < total;
       i += gridDim.x * blockDim.x) {
    int f = i & (NODE_F - 1);
    float v = x[i] + nodeacc[i] * s2[f] + t2[f];
    x[i] = softplusf_(v);
    nodeacc[i] = 0.0f;
    if (f == 0) ncount[i >> 6] = 0.0f;
  }
}

// ---------------------------------------------------------------- misc small kernels
__global__ void cg_zero(float* __restrict__ p, int total) {
  for (int i = blockIdx.x * blockDim.x + threadIdx.x; i < total;
       i += gridDim.x * blockDim.x) p[i] = 0.0f;
}

__global__ void cg_embed(const int* __restrict__ nf, const float* __restrict__ emb,
                         float* __restrict__ x, int total) {
  for (int i = blockIdx.x * blockDim.x + threadIdx.x; i < total;
       i += gridDim.x * blockDim.x) {
    int n = i >> 6, f = i & (NODE_F - 1);
    x[i] = emb[(size_t)nf[n] * NODE_F + f];
  }
}

__global__ void cg_crys_acc(const float* __restrict__ x, const int* __restrict__ idx3,
                            float* __restrict__ crysacc, float* __restrict__ ccount,
                            int total) {
  for (int i = blockIdx.x * blockDim.x + threadIdx.x; i < total;
       i += gridDim.x * blockDim.x) {
    int n = i >> 6, f = i & (NODE_F - 1);
    int c = idx3[n];
    atomicAdd(&crysacc[(size_t)c * NODE_F + f], x[i]);
    if (f == 0) atomicAdd(&ccount[c], 1.0f);
  }
}

__global__ void __launch_bounds__(128) cg_crys_out(
    const float* __restrict__ crysacc, const float* __restrict__ ccount,
    const float* __restrict__ Wc, const float* __restrict__ bc,
    const float* __restrict__ Wo, const float* __restrict__ bo,
    float* __restrict__ out) {
  int c = blockIdx.x, tid = threadIdx.x;
  __shared__ float m[NODE_F];
  __shared__ float r0[128];
  __shared__ float r1[128];
  if (tid < NODE_F) {
    float cnt = fmaxf(ccount[c], 1.0f);
    m[tid] = crysacc[(size_t)c * NODE_F + tid] / cnt;
  }
  __syncthreads();
  float h = bc[tid];
  for (int k = 0; k < NODE_F; ++k) h += m[k] * Wc[(size_t)k * 128 + tid];
  h = softplusf_(h);
  r0[tid] = h * Wo[tid * 2 + 0];
  r1[tid] = h * Wo[tid * 2 + 1];
  __syncthreads();
  for (int s = 64; s > 0; s >>= 1) {
    if (tid < s) { r0[tid] += r0[tid + s]; r1[tid] += r1[tid + s]; }
    __syncthreads();
  }
  if (tid == 0) {
    out[c * 2 + 0] = r0[0] + bo[0];
    out[c * 2 + 1] = r1[0] + bo[1];
  }
}

// ---------------------------------------------------------------- launch
extern "C" void kernel_launch(void* const* d_in, const int* in_sizes, int n_in,
                              void* d_out, int out_size, void* d_ws, size_t ws_size,
                              hipStream_t stream) {
  const int*   node_fea = (const int*)  d_in[0];
  const float* edge_fea = (const float*)d_in[1];
  const int*   idx1     = (const int*)  d_in[2];
  const int*   idx2     = (const int*)  d_in[3];
  const int*   idx3     = (const int*)  d_in[4];
  const float* emb      = (const float*)d_in[5];
  const float* Wf       = (const float*)d_in[6];
  const float* bf       = (const float*)d_in[7];
  const float* g1       = (const float*)d_in[8];
  const float* be1      = (const float*)d_in[9];
  const float* g2       = (const float*)d_in[10];
  const float* be2      = (const float*)d_in[11];
  const float* Wc       = (const float*)d_in[12];
  const float* bc       = (const float*)d_in[13];
  const float* Wo       = (const float*)d_in[14];
  const float* bo       = (const float*)d_in[15];
  float* out = (float*)d_out;

  const int N = in_sizes[0];            // 100000 nodes
  const int E = in_sizes[1] / EDGE_F;   // 1.6M edges
  const int C = out_size / 2;           // 2000 crystals
  const int ntiles = E / 16;            // E divisible by 16 for this model

  // workspace layout (floats)
  float* ws       = (float*)d_ws;
  float* x        = ws;                         // N*64
  float* nodeacc  = x + (size_t)N * NODE_F;     // N*64
  float* ncount   = nodeacc + (size_t)N * NODE_F; // N
  float* crysacc  = ncount + N;                 // C*64
  float* ccount   = crysacc + (size_t)C * NODE_F; // C
  float* zsum     = ccount + C;                 // 128
  float* zsq      = zsum + 128;                 // 128
  float* s1       = zsq + 128;                  // 128
  float* t1       = s1 + 128;                   // 128
  float* nsum     = t1 + 128;                   // 64
  float* nsq      = nsum + 64;                  // 64
  float* s2       = nsq + 64;                   // 64
  float* t2       = s2 + 64;                    // 64
  float2* wswz    = (float2*)(t2 + 64);         // 3 * 11008 float2

  const int zero_total = (int)((size_t)N * NODE_F + N + (size_t)C * NODE_F + C + 768);

  const int GEMM_BLOCKS = 2048;
  const int EW_BLOCKS   = 4096;

  cg_zero<<<EW_BLOCKS, 256, 0, stream>>>(nodeacc, zero_total);
  cg_prep_w<<<(3 * FRAGS_PER_LAYER + 127) / 128, 128, 0, stream>>>(Wf, wswz);
  cg_embed<<<EW_BLOCKS, 256, 0, stream>>>(node_fea, emb, x, N * NODE_F);

  for (int l = 0; l < 3; ++l) {
    const float2* wl  = wswz + (size_t)l * FRAGS_PER_LAYER;
    const float*  bfl = bf  + (size_t)l * DOUT;

    cg_gemm_stats<<<GEMM_BLOCKS, 128, 0, stream>>>(
        x, edge_fea, idx1, idx2, wl, bfl, ntiles, zsum, zsq);
    cg_bn_final<<<1, 128, 0, stream>>>(
        zsum, zsq, g1 + (size_t)l * DOUT, be1 + (size_t)l * DOUT,
        1.0f / (float)E, s1, t1, DOUT);
    cg_msg_scatter<<<GEMM_BLOCKS, 128, 0, stream>>>(
        x, edge_fea, idx1, idx2, wl, bfl, s1, t1, nodeacc, ncount, ntiles);
    cg_node_stats<<<EW_BLOCKS, 256, 0, stream>>>(
        nodeacc, ncount, N * NODE_F, nsum, nsq);
    cg_bn_final<<<1, 64, 0, stream>>>(
        nsum, nsq, g2 + (size_t)l * NODE_F, be2 + (size_t)l * NODE_F,
        1.0f / (float)N, s2, t2, NODE_F);
    cg_node_update<<<EW_BLOCKS, 256, 0, stream>>>(
        x, nodeacc, ncount, s2, t2, N * NODE_F);
  }

  cg_crys_acc<<<EW_BLOCKS, 256, 0, stream>>>(x, idx3, crysacc, ccount, N * NODE_F);
  cg_crys_out<<<C, 128, 0, stream>>>(crysacc, ccount, Wc, bc, Wo, bo, out);
}